// GAFN_19292993093863
// MI455X (gfx1250) — compile-verified
//
#include <hip/hip_runtime.h>
#include <math.h>

typedef __attribute__((ext_vector_type(2))) float v2f;
typedef __attribute__((ext_vector_type(8))) float v8f;

#define NCH 8      // channels
#define DIM 128    // descriptor dim
#define KCL 64     // clusters
#define ROWS 16    // 2 descriptors * 8 channels per block

// ---------------------------------------------------------------------------
// Kernel 1: normalize x, WMMA logits, softmax, residual + intra-norm,
//           per-(n,c) sum/sumsq (P,Q) and per-channel sums (S1,S2).
// Block: 128 threads (4 waves). Each block owns 2 descriptors (16 rows).
// Wave w owns cluster tile [16w, 16w+16).
// ---------------------------------------------------------------------------
__global__ __launch_bounds__(128)
void netvlad_k1(const float* __restrict__ x,
                const float* __restrict__ conv_w,
                const float* __restrict__ conv_b,
                const float* __restrict__ centroids,
                float* __restrict__ out,
                float* __restrict__ ws,
                int N)
{
    __shared__ float xh[ROWS][DIM];    // normalized descriptors (8 KB)
    __shared__ float lg[ROWS][KCL];    // logits -> soft_assign (4 KB)
    __shared__ float red[ROWS][8];     // row reduction scratch
    __shared__ float rowP[ROWS], rowQ[ROWS];

    const int t    = threadIdx.x;          // 0..127
    const int wave = t >> 5;
    const int lane = t & 31;
    const int n0   = blockIdx.x * 2;       // first of two descriptors

    // ---- load x rows + squared-sum: 8 threads per row, 16 elems each ----
    const int r   = t >> 3;                // row 0..15  -> (n0 + r/8, c = r%8)
    const int sub = t & 7;
    const float* xrow = x + ((size_t)n0 * NCH + r) * DIM;
    float ss = 0.f;
    #pragma unroll
    for (int i = 0; i < 16; ++i) {
        float v = xrow[sub * 16 + i];
        xh[r][sub * 16 + i] = v;
        ss += v * v;
    }
    red[r][sub] = ss;
    if (t < ROWS) { rowP[t] = 0.f; rowQ[t] = 0.f; }
    __syncthreads();
    if (sub == 0) {
        float s = 0.f;
        #pragma unroll
        for (int i = 0; i < 8; ++i) s += red[r][i];
        red[r][0] = 1.0f / fmaxf(sqrtf(s), 1e-12f);
    }
    __syncthreads();
    const float invn = red[r][0];
    #pragma unroll
    for (int i = 0; i < 16; ++i) xh[r][sub * 16 + i] *= invn;
    __syncthreads();

    // ---- WMMA logits: 16 rows x 16 clusters per wave, K = 128 in steps of 4
    // A 16x4 f32 layout: lanes 0-15 K={0,1}, lanes 16-31 K={2,3}
    const int colbase = wave * 16;
    const int m    = lane & 15;            // A row / B column within tile
    const int koff = (lane >> 4) * 2;      // K sub-offset for this half-wave
    const float* wrow = conv_w + (size_t)(colbase + m) * DIM;
    v8f acc = {};
    #pragma unroll
    for (int kb = 0; kb < DIM; kb += 4) {
        v2f a, b;
        a.x = xh[m][kb + koff];
        a.y = xh[m][kb + koff + 1];
        b.x = wrow[kb + koff];
        b.y = wrow[kb + koff + 1];
        acc = __builtin_amdgcn_wmma_f32_16x16x4_f32(
                  false, a, false, b, (short)0, acc, false, false);
    }
    // C/D layout: vgpr v, lanes 0-15 -> M=v, lanes 16-31 -> M=v+8
    {
        const int ncol = lane & 15;
        const int mofs = (lane >> 4) * 8;
        const float bias = conv_b[colbase + ncol];
        #pragma unroll
        for (int v = 0; v < 8; ++v)
            lg[mofs + v][colbase + ncol] = acc[v] + bias;
    }
    __syncthreads();

    // ---- softmax over 64 clusters per row (axis=1 of (N,K,C)) ----
    if (sub == 0) {
        float mx = lg[r][0];
        for (int i = 1; i < KCL; ++i) mx = fmaxf(mx, lg[r][i]);
        float s = 0.f;
        for (int i = 0; i < KCL; ++i) {
            float e = expf(lg[r][i] - mx);
            lg[r][i] = e;
            s += e;
        }
        red[r][0] = 1.0f / s;
    }
    __syncthreads();
    {
        const float isum = red[r][0];
        #pragma unroll
        for (int i = 0; i < 8; ++i) lg[r][sub * 8 + i] *= isum;
    }
    __syncthreads();

    // ---- residual + intra-norm; each wave: 16 rows x its 16 clusters ----
    for (int rr = 0; rr < ROWS; ++rr) {
        const int n = n0 + (rr >> 3);
        const int c = rr & 7;
        for (int kk = 0; kk < 16; ++kk) {
            const int k = colbase + kk;
            const float a = lg[rr][k];
            const float* cent = centroids + (size_t)k * DIM;
            float v0[4];
            float ssq = 0.f;
            #pragma unroll
            for (int j = 0; j < 4; ++j) {
                int d = lane + 32 * j;
                float rv = (xh[rr][d] - cent[d]) * a;
                v0[j] = rv;
                ssq += rv * rv;
            }
            #pragma unroll
            for (int off = 16; off >= 1; off >>= 1)
                ssq += __shfl_xor(ssq, off, 32);
            const float scale = 1.0f / fmaxf(sqrtf(ssq), 1e-12f);
            float psum = 0.f, psq = 0.f;
            float* orow = out + (((size_t)n * NCH + c) * KCL + k) * DIM;
            #pragma unroll
            for (int j = 0; j < 4; ++j) {
                float w = v0[j] * scale;
                orow[lane + 32 * j] = w;
                psum += w;
                psq  += w * w;
            }
            #pragma unroll
            for (int off = 16; off >= 1; off >>= 1) {
                psum += __shfl_xor(psum, off, 32);
                psq  += __shfl_xor(psq,  off, 32);
            }
            if (lane == 0) {
                atomicAdd(&rowP[rr], psum);   // LDS f32 atomic
                atomicAdd(&rowQ[rr], psq);
            }
        }
    }
    __syncthreads();

    // ---- publish per-(n,c) P,Q (owned) and per-channel sums (atomic) ----
    if (t < ROWS) {
        const int n = n0 + (t >> 3);
        const int c = t & 7;
        const size_t nc = (size_t)n * NCH + c;
        ws[16 + nc] = rowP[t];                            // P[n,c]
        ws[16 + (size_t)N * NCH + nc] = rowQ[t];          // Q[n,c]
        atomicAdd(&ws[c], rowP[t]);                       // S1[c]
        atomicAdd(&ws[8 + c], rowQ[t]);                   // S2[c]
    }
}

// ---------------------------------------------------------------------------
// Kernel 2: fold BatchNorm (batch stats) + final L2 norm, rewrite in place.
// One block per (n,c). y = s*v + t ; ||y||^2 = s^2*Q + 2*s*t*P + 8192*t^2.
// ---------------------------------------------------------------------------
__global__ __launch_bounds__(256)
void netvlad_k2(float* __restrict__ out,
                const float* __restrict__ ws,
                const float* __restrict__ gamma,
                const float* __restrict__ beta,
                int N)
{
    const int nc = blockIdx.x;             // 0 .. N*8-1
    const int c  = nc & 7;
    const float cnt  = (float)N * (float)(KCL * DIM);
    const float S1   = ws[c];
    const float S2   = ws[8 + c];
    const float mean = S1 / cnt;
    const float var  = S2 / cnt - mean * mean;
    const float istd = 1.0f / sqrtf(var + 1e-5f);
    const float s    = gamma[c] * istd;
    const float tt   = beta[c] - mean * s;

    const float P = ws[16 + nc];
    const float Q = ws[16 + (size_t)N * NCH + nc];
    const float normsq = s * s * Q + 2.0f * s * tt * P
                       + (float)(KCL * DIM) * tt * tt;
    const float inv = 1.0f / fmaxf(sqrtf(normsq), 1e-12f);

    float4* row4 = reinterpret_cast<float4*>(out + (size_t)nc * (KCL * DIM));
    for (int i = threadIdx.x; i < (KCL * DIM) / 4; i += blockDim.x) {
        float4 v = row4[i];
        v.x = (v.x * s + tt) * inv;
        v.y = (v.y * s + tt) * inv;
        v.z = (v.z * s + tt) * inv;
        v.w = (v.w * s + tt) * inv;
        row4[i] = v;
    }
}

// ---------------------------------------------------------------------------
extern "C" void kernel_launch(void* const* d_in, const int* in_sizes, int n_in,
                              void* d_out, int out_size, void* d_ws, size_t ws_size,
                              hipStream_t stream)
{
    const float* x         = (const float*)d_in[0];
    const float* conv_w    = (const float*)d_in[1];
    const float* conv_b    = (const float*)d_in[2];
    const float* centroids = (const float*)d_in[3];
    const float* bn_gamma  = (const float*)d_in[4];
    const float* bn_beta   = (const float*)d_in[5];

    float* out = (float*)d_out;
    float* ws  = (float*)d_ws;

    const int N = in_sizes[0] / (NCH * DIM);   // 2048

    // zero the 16 per-channel accumulators (graph-capture safe)
    hipMemsetAsync(ws, 0, 16 * sizeof(float), stream);

    netvlad_k1<<<N / 2, 128, 0, stream>>>(x, conv_w, conv_b, centroids,
                                          out, ws, N);
    netvlad_k2<<<N * NCH, 256, 0, stream>>>(out, ws, bn_gamma, bn_beta, N);
}